// AdExternal_N3Tree_43473658970709
// MI455X (gfx1250) — compile-verified
//
#include <hip/hip_runtime.h>
#include <math.h>

// MI455X / gfx1250, wave32.
// Phase A: independent per-parent chains of <=8 sequential 32x256 matvecs (VALU,
//          LDS-broadcast block), feats via ds_add_f32 -> global_atomic_add_f32.
// Phase B: leaf gather + fused dual-MLP using V_WMMA_F32_16X16X4_F32 (fp32 WMMA:
//          keeps reference precision; problem is bandwidth/latency bound at ~90MB,
//          ~4GFLOP, so low-precision WMMA would gain nothing).

typedef __attribute__((ext_vector_type(2))) float v2f;
typedef __attribute__((ext_vector_type(8))) float v8f;

__global__ void init_feats_kernel(float* feats) {
    if (threadIdx.x < 32) feats[threadIdx.x] = 0.0f;
}

// ---------------- Phase A: tree-conv scan over independent chains ----------------
__global__ __launch_bounds__(256) void tree_scan_kernel(
    float* __restrict__ dataw,            // [M*256] working copy (mutated)
    const float* __restrict__ conv_w,     // [10*32*32*8]
    const float* __restrict__ conv_b,     // [10*32]
    const float* __restrict__ depth_weight, // [10]
    const int* __restrict__ idx_sorted,   // [M] descending
    const int* __restrict__ depth_sorted, // [M] descending
    const int* __restrict__ node_depth,   // [M]
    float* __restrict__ feats_g,          // [32] global accumulator
    int M)
{
    __shared__ float blk[8][256];   // one 8x32 block per wave
    __shared__ float facc_s[32];

    const int lane = threadIdx.x & 31;   // = output channel o = feature index
    const int wid  = threadIdx.x >> 5;
    const int gw   = blockIdx.x * 8 + wid;   // sorted position this wave owns

    if (threadIdx.x < 32) facc_s[threadIdx.x] = 0.0f;
    __syncthreads();

    float facc = 0.0f;
    if (gw < M) {
        const int idx0 = idx_sorted[gw];
        const int p = idx0 >> 3;
        const bool isStart = (gw == 0) || ((idx_sorted[gw - 1] >> 3) != p);
        if (isStart) {
            float* myblk = &blk[wid][0];
            const int pb = p * 256;
            #pragma unroll
            for (int k = 0; k < 8; ++k)                     // coalesced block load
                myblk[k * 32 + lane] = dataw[pb + k * 32 + lane];

            const int cd = node_depth[p];                   // constant along chain
            const float* Wb  = conv_w + cd * 8192 + lane * 256; // row for o=lane
            const float bias = conv_b[cd * 32 + lane];

            int j = gw;
            while (j < M && (idx_sorted[j] >> 3) == p) {    // sequential chain (<=~8)
                const int c = idx_sorted[j] & 7;
                float acc = bias;
                // feat[o] = sum_{i,k} W[o][i][k]*blk[k][i]; W row contiguous (i*8+k)
                #pragma unroll 8
                for (int q = 0; q < 256; q += 4) {
                    const float4 w4 = *reinterpret_cast<const float4*>(Wb + q);
                    const int i = q >> 3;   // feature idx (constant within group of 8)
                    const int k = q & 7;    // cell idx base (0 or 4)
                    // all lanes read the SAME LDS address -> broadcast, no conflicts
                    acc = fmaf(w4.x, myblk[(k + 0) * 32 + i], acc);
                    acc = fmaf(w4.y, myblk[(k + 1) * 32 + i], acc);
                    acc = fmaf(w4.z, myblk[(k + 2) * 32 + i], acc);
                    acc = fmaf(w4.w, myblk[(k + 3) * 32 + i], acc);
                }
                facc = fmaf(depth_weight[depth_sorted[j]], acc, facc);
                myblk[c * 32 + lane] = acc;        // in-chain RAW via LDS
                dataw[pb + c * 32 + lane] = acc;   // final state for leaf gather
                ++j;
            }
        }
    }
    atomicAdd(&facc_s[lane], facc);                // ds_add_f32 (block-level)
    __syncthreads();
    if (threadIdx.x < 32)
        atomicAdd(&feats_g[threadIdx.x], facc_s[threadIdx.x]); // global_atomic_add_f32
}

// ---------------- Phase B: leaf gather + fused MLPs via fp32 WMMA ----------------
__global__ __launch_bounds__(128) void leaf_mlp_kernel(
    const float* __restrict__ dataw,   // [M*256] post-scan tree data
    const float* __restrict__ feats_g, // [32]
    const float* __restrict__ hf_w1, const float* __restrict__ hf_b1,
    const float* __restrict__ hf_w2, const float* __restrict__ hf_b2,
    const float* __restrict__ hs_w1, const float* __restrict__ hs_b1,
    const float* __restrict__ hs_w2, const float* __restrict__ hs_b2,
    const int* __restrict__ leaf_idx,  // [L]
    float* __restrict__ out,           // [L*4]
    int L)
{
    __shared__ float W1c[32 * 128];    // fused [hf_w1 | hs_w1]
    __shared__ float b1c[128];
    __shared__ float W2c[128 * 4];     // block-diagonal fused second layer
    __shared__ float b2c[4];
    __shared__ float hbuf[4][16 * 128]; // per-wave GELU'd hidden tile

    const int tid = threadIdx.x;
    for (int idx = tid; idx < 32 * 128; idx += 128) {
        const int r = idx >> 7, col = idx & 127;
        W1c[idx] = (col < 64) ? hf_w1[r * 64 + col] : hs_w1[r * 64 + (col - 64)];
    }
    if (tid < 128) b1c[tid] = (tid < 64) ? hf_b1[tid] : hs_b1[tid - 64];
    for (int idx = tid; idx < 512; idx += 128) {
        const int hid = idx >> 2, j = idx & 3;
        float v = 0.0f;
        if (hid < 64) { if (j < 3) v = hf_w2[hid * 3 + j]; }
        else          { if (j == 3) v = hs_w2[hid - 64];   }
        W2c[idx] = v;
    }
    if (tid < 4) b2c[tid] = (tid < 3) ? hf_b2[tid] : hs_b2[0];
    __syncthreads();

    const int lane = tid & 31;
    const int wid  = tid >> 5;
    const int lm   = lane & 15;            // A row / B col within tile
    const int koff = (lane >> 4) * 2;      // K sub-offset per 16x16x4 layout
    const int base = (blockIdx.x * 4 + wid) * 16;   // first leaf of this wave's tile

    // A fragments: x[row][4s+koff .. +1], row = leaf base+lm, x = data[leaf]+feats
    const int  row   = base + lm;
    const bool valid = row < L;
    const int  cellbase = valid ? leaf_idx[row] * 32 : 0;
    v2f afrag[8];
    #pragma unroll
    for (int s = 0; s < 8; ++s) {
        const int col = 4 * s + koff;
        float x0 = 0.0f, x1 = 0.0f;
        if (valid) {
            x0 = dataw[cellbase + col]     + feats_g[col];
            x1 = dataw[cellbase + col + 1] + feats_g[col + 1];
        }
        afrag[s].x = x0; afrag[s].y = x1;
    }

    float* h = &hbuf[wid][0];
    const int rbase = (lane >> 4) * 8;     // C/D rows held by this lane
    #pragma unroll
    for (int t = 0; t < 8; ++t) {          // 8 hidden N-tiles of 16
        const float bias = b1c[16 * t + lm];
        v8f c = { bias, bias, bias, bias, bias, bias, bias, bias };
        #pragma unroll
        for (int s = 0; s < 8; ++s) {      // K=32 in steps of 4
            v2f b;
            b.x = W1c[(4 * s + koff)     * 128 + 16 * t + lm];
            b.y = W1c[(4 * s + koff + 1) * 128 + 16 * t + lm];
            c = __builtin_amdgcn_wmma_f32_16x16x4_f32(
                    false, afrag[s], false, b, (short)0, c, false, false);
        }
        #pragma unroll
        for (int r = 0; r < 8; ++r) {      // exact GELU, stage to LDS
            const float x = c[r];
            const float g = 0.5f * x * (1.0f + erff(x * 0.70710678118654752f));
            h[(rbase + r) * 128 + 16 * t + lm] = g;
        }
    }

    // Second layer: 64 outputs (16 leaves x 4); each lane computes 2.
    #pragma unroll
    for (int oo = 0; oo < 2; ++oo) {
        const int o = lane * 2 + oo;
        const int m = o >> 2, j = o & 3;
        float acc = b2c[j];
        const float* hm = &h[m * 128];
        #pragma unroll 8
        for (int hid = 0; hid < 128; ++hid)
            acc = fmaf(hm[hid], W2c[hid * 4 + j], acc);
        const int gl = base + m;
        if (gl < L) out[gl * 4 + j] = acc;
    }
}

extern "C" void kernel_launch(void* const* d_in, const int* in_sizes, int n_in,
                              void* d_out, int out_size, void* d_ws, size_t ws_size,
                              hipStream_t stream)
{
    const float* data         = (const float*)d_in[0];
    const float* conv_w       = (const float*)d_in[1];
    const float* conv_b       = (const float*)d_in[2];
    const float* depth_weight = (const float*)d_in[3];
    const float* hf_w1 = (const float*)d_in[4];
    const float* hf_b1 = (const float*)d_in[5];
    const float* hf_w2 = (const float*)d_in[6];
    const float* hf_b2 = (const float*)d_in[7];
    const float* hs_w1 = (const float*)d_in[8];
    const float* hs_b1 = (const float*)d_in[9];
    const float* hs_w2 = (const float*)d_in[10];
    const float* hs_b2 = (const float*)d_in[11];
    const int* idx_sorted   = (const int*)d_in[12];
    const int* depth_sorted = (const int*)d_in[13];
    const int* node_depth   = (const int*)d_in[14];
    const int* leaf_idx     = (const int*)d_in[15];

    const int M = in_sizes[12];   // 30000
    const int L = in_sizes[15];   // 7*M+1

    float* feats_g = (float*)d_ws;
    float* dataw   = (float*)((char*)d_ws + 256);
    const size_t dbytes = (size_t)M * 256 * sizeof(float);
    if (ws_size < dbytes + 256) return;   // deterministic given fixed sizes

    // working copy: the scan scatters into the tree; inputs must stay intact
    hipMemcpyAsync(dataw, data, dbytes, hipMemcpyDeviceToDevice, stream);
    init_feats_kernel<<<1, 32, 0, stream>>>(feats_g);

    const int blocksA = (M + 7) / 8;      // one wave per sorted position
    tree_scan_kernel<<<blocksA, 256, 0, stream>>>(
        dataw, conv_w, conv_b, depth_weight,
        idx_sorted, depth_sorted, node_depth, feats_g, M);

    const int tiles   = (L + 15) / 16;    // 16 leaves per wave
    const int blocksB = (tiles + 3) / 4;  // 4 waves per block
    leaf_mlp_kernel<<<blocksB, 128, 0, stream>>>(
        dataw, feats_g, hf_w1, hf_b1, hf_w2, hf_b2,
        hs_w1, hs_b1, hs_w2, hs_b2, leaf_idx, (float*)d_out, L);
}